// MNB_8151847928093
// MI455X (gfx1250) — compile-verified
//
#include <hip/hip_runtime.h>
#include <hip/hip_bf16.h>
#include <stdint.h>

// Multinomial-NB-style presence featurizer + linear head.
// y[doc] = sum_{unique tokens t in doc} W[t] + bias
//
// Roofline: sparse gather (~3.5 MB traffic) beats dense one-hot GEMM (>=205 MB)
// by ~60x, so the kernel is HBM/latency bound (~0.15 us @ 23.3 TB/s).
//
// Per-document workgroup (4096 blocks x 256 threads = 8 wave32s):
//   1) global_load_async_to_lds_b32 stages the 200 strided token ids into LDS
//      (ASYNCcnt), overlapped with zeroing a 50000-bit LDS presence bitset
//   2) s_wait_asynccnt 0 + barrier
//   3) dedup via ds_or_rtn_b32; first setter of a bit gathers W[t]
//   4) block reduction via V_WMMA_F32_16X16X4_F32 against an all-ones B:
//      with B==1, D[m][n] = sum_k A[m][k] + C[m][n], so
//      sum(all 256 elems of D) == 16 * sum(all 64 elems of A) regardless of
//      the VGPR<->element mapping. Four chained WMMAs reduce 256 partials;
//      wave 0 finishes with per-lane adds + __shfl_down, scales by 1/16.

#define SEQ_LEN  200
#define BATCH    4096
#define VOCAB    50000
#define BITWORDS 1568          // ceil(50000/32)=1563, padded to a multiple of 32
#define THREADS  256           // 8 waves (wave32)

typedef __attribute__((address_space(3))) int lds_int;
typedef __attribute__((ext_vector_type(2))) float v2f;
typedef __attribute__((ext_vector_type(8))) float v8f;

__global__ __launch_bounds__(THREADS) void mnb_presence_linear(
    const int*   __restrict__ text,   // [SEQ_LEN, BATCH] token ids (int32)
    const float* __restrict__ W,      // [VOCAB]
    const float* __restrict__ bias,   // [1]
    float*       __restrict__ out)    // [BATCH]
{
    __shared__ unsigned bitset[BITWORDS];
    __shared__ int      toks[SEQ_LEN];
    __shared__ float    red[THREADS];

    const int doc = blockIdx.x;
    const int tid = threadIdx.x;

    // ---- Stage 1: async global->LDS copy of this doc's tokens (16 KB stride
    // between consecutive l). Flies on ASYNCcnt while we zero the bitset.
    if (tid < SEQ_LEN) {
        const int* gaddr = text + (size_t)tid * BATCH + doc;
        unsigned lds_off = (unsigned)(uintptr_t)(lds_int*)&toks[tid];
        asm volatile("global_load_async_to_lds_b32 %0, %1, off"
                     :: "v"(lds_off), "v"(gaddr)
                     : "memory");
    }

    // ---- Overlapped: zero the presence bitset (1568 words / 256 threads).
    for (int i = tid; i < BITWORDS; i += THREADS) bitset[i] = 0u;

    // ---- Close the async pipe; make all LDS writes visible.
    asm volatile("s_wait_asynccnt 0x0" ::: "memory");
    __syncthreads();

    // ---- Stage 2: dedup + gather. ds_or_rtn_b32 returns the pre-op word, so
    // exactly one thread per unique token sees its bit previously clear.
    float local = 0.0f;
    if (tid < SEQ_LEN) {
        const int t        = toks[tid];        // 0 <= t < VOCAB
        const unsigned bit = 1u << (t & 31);
        const unsigned old = atomicOr(&bitset[(unsigned)t >> 5], bit);
        if ((old & bit) == 0u) local = W[t];   // first occurrence only
    }

    red[tid] = local;
    __syncthreads();

    // ---- Stage 3: wave-0 WMMA reduction of the 256 partial sums.
    // tid<32 is exactly wave 0 under wave32, so EXEC is all-ones inside
    // (WMMA requirement). Chunk c packs red[c*64 .. c*64+63] into the two
    // A VGPRs (a.x = lane, a.y = lane+32) -- any bijection is valid since
    // B is all-ones (see header comment).
    if (tid < 32) {
        const v2f ones = {1.0f, 1.0f};
        v8f acc = {};
        #pragma unroll
        for (int c = 0; c < 4; ++c) {
            v2f a;
            a.x = red[c * 64 + tid];
            a.y = red[c * 64 + 32 + tid];
            acc = __builtin_amdgcn_wmma_f32_16x16x4_f32(
                /*neg_a=*/false, a, /*neg_b=*/false, ones,
                /*c_mod=*/(short)0, acc, /*reuse_a=*/false, /*reuse_b=*/false);
        }
        // Per-lane sum of the 8 accumulator VGPRs == this lane's share of
        // sum(all D elements).
        float s = ((acc[0] + acc[1]) + (acc[2] + acc[3])) +
                  ((acc[4] + acc[5]) + (acc[6] + acc[7]));
        // Cross-lane sum over the 32 lanes of wave 0.
        #pragma unroll
        for (int off = 16; off > 0; off >>= 1)
            s += __shfl_down(s, off, 32);
        if (tid == 0)
            out[doc] = s * 0.0625f + bias[0];   // /16 is exact (pow2)
    }
}

extern "C" void kernel_launch(void* const* d_in, const int* in_sizes, int n_in,
                              void* d_out, int out_size, void* d_ws, size_t ws_size,
                              hipStream_t stream) {
    const int*   text = (const int*)  d_in[0];  // [200*4096] token ids
    const float* W    = (const float*)d_in[1];  // [50000]
    const float* b    = (const float*)d_in[2];  // [1]
    float*       out  = (float*)      d_out;    // [4096]
    (void)in_sizes; (void)n_in; (void)out_size; (void)d_ws; (void)ws_size;

    mnb_presence_linear<<<BATCH, THREADS, 0, stream>>>(text, W, b, out);
}